// GCNModel_18330920419456
// MI455X (gfx1250) — compile-verified
//
#include <hip/hip_runtime.h>
#include <hip/hip_bf16.h>

typedef float v2f __attribute__((ext_vector_type(2)));
typedef float v8f __attribute__((ext_vector_type(8)));

#define HIDF 256
#define NCLS 3

// ---------------- elementwise / setup kernels ----------------

__global__ void fill_kernel(float* p, float val, int n) {
  int i = blockIdx.x * blockDim.x + threadIdx.x;
  if (i < n) p[i] = val;
}

__global__ void deg_count_kernel(const int* __restrict__ dst, float* __restrict__ deg, int nE) {
  int e = blockIdx.x * blockDim.x + threadIdx.x;
  if (e < nE) atomicAdd(&deg[dst[e]], 1.0f);
}

__global__ void deg_rsqrt_kernel(float* deg, int n) {
  int i = blockIdx.x * blockDim.x + threadIdx.x;
  if (i < n) {
    float d = deg[i];
    deg[i] = d > 0.0f ? rsqrtf(d) : 0.0f;
  }
}

// agg[i,f] = xw[i,f] * dinv[i]^2 + bias[f]   (self-loop term + bias)
__global__ void agg_init_kernel(const float* __restrict__ xw, const float* __restrict__ dinv,
                                const float* __restrict__ bias, float* __restrict__ agg, int n) {
  int i = blockIdx.x * blockDim.x + threadIdx.x;
  if (i < n) {
    int node = i >> 8;          // HIDF == 256
    int f = i & (HIDF - 1);
    float di = dinv[node];
    agg[i] = xw[i] * di * di + bias[f];
  }
}

// one wave (32 lanes) per edge; each lane handles 8 of 256 features
__global__ void edge_scatter_kernel(const float* __restrict__ xw, const float* __restrict__ dinv,
                                    const int* __restrict__ src, const int* __restrict__ dst,
                                    float* __restrict__ agg, int nE) {
  int e = blockIdx.x * (blockDim.x >> 5) + (threadIdx.x >> 5);
  if (e >= nE) return;
  int lane = threadIdx.x & 31;
  int s = src[e];
  int d = dst[e];
  float c = dinv[s] * dinv[d];
  const float* xs = xw + (size_t)s * HIDF;
  float* ad = agg + (size_t)d * HIDF;
  #pragma unroll
  for (int f = lane; f < HIDF; f += 32) {
    atomicAdd(&ad[f], xs[f] * c);
  }
}

// one wave per node: accumulate relu(h) into per-graph sums + counts
__global__ void pool_acc_kernel(const float* __restrict__ h, const int* __restrict__ batch,
                                float* __restrict__ sums, float* __restrict__ cnt, int nNodes) {
  int node = blockIdx.x * (blockDim.x >> 5) + (threadIdx.x >> 5);
  if (node >= nNodes) return;
  int lane = threadIdx.x & 31;
  int g = batch[node];
  const float* hr = h + (size_t)node * HIDF;
  float* sg = sums + (size_t)g * HIDF;
  #pragma unroll
  for (int f = lane; f < HIDF; f += 32) {
    atomicAdd(&sg[f], fmaxf(hr[f], 0.0f));   // fused layer-2 ReLU
  }
  if (lane == 0) atomicAdd(&cnt[g], 1.0f);
}

__global__ void pool_div_kernel(float* sums, const float* cnt, int n) {
  int i = blockIdx.x * blockDim.x + threadIdx.x;
  if (i < n) {
    int g = i >> 8;
    sums[i] = sums[i] / fmaxf(cnt[g], 1.0f);
  }
}

// tiny head: out[g,c] = sum_k h[g,k]*W[k,c] + b[c]
__global__ void fc2_kernel(const float* __restrict__ h, const float* __restrict__ W,
                           const float* __restrict__ b, float* __restrict__ out, int G) {
  int i = blockIdx.x * blockDim.x + threadIdx.x;
  if (i >= G * NCLS) return;
  int g = i / NCLS;
  int c = i - g * NCLS;
  const float* hr = h + (size_t)g * HIDF;
  float acc = b[c];
  #pragma unroll 8
  for (int k = 0; k < HIDF; ++k) acc = fmaf(hr[k], W[k * NCLS + c], acc);
  out[i] = acc;
}

// ---------------- f32 WMMA GEMM, strip-mined ----------------
// C[M,N] = op(A)[M,K] @ B[K,N] (+bias[col]) (optional relu out), row-major f32.
// op(A) = relu(A) if RELU_IN. Requires M%16==0, N%64==0, K%4==0.
// One wave computes a 16x64 strip = 4 N-tiles sharing the A fragment
// (4x less HBM traffic on A than one-tile-per-wave).
//
// V_WMMA_F32_16X16X4_F32 lane layouts (ISA 7.12.2):
//   A 16x4: lane m=(l&15), kh=(l>>4); VGPR v holds A[m][2*kh+v]
//   B 4x16: lane n=(l&15), kh=(l>>4); VGPR v holds B[2*kh+v][n]
//   C/D 16x16: lane n=(l&15); VGPR v holds C[v + 8*(l>>4)][n]
template <bool RELU_IN, bool RELU_OUT, bool BIAS>
__global__ void wmma_gemm_f32_strip_kernel(const float* __restrict__ A,
                                           const float* __restrict__ B,
                                           const float* __restrict__ bias,
                                           float* __restrict__ C,
                                           int M, int N, int K) {
  const int lane = threadIdx.x & 31;
  const int wave = threadIdx.x >> 5;
  const int nstrips = N >> 6;                 // strips of 64 columns
  const int strips = (M >> 4) * nstrips;
  int sid = blockIdx.x * (blockDim.x >> 5) + wave;
  if (sid >= strips) return;                  // wave-uniform: EXEC all-1s for WMMA
  const int mt = sid / nstrips;
  const int ns = sid - mt * nstrips;
  const int mn = lane & 15;                   // A-row / B,C,D-column within tile
  const int kh = lane >> 4;                   // K-half selector

  const float* arow  = A + (size_t)(mt * 16 + mn) * K + 2 * kh;
  const float* bbase = B + (size_t)(2 * kh) * N + ns * 64 + mn;

  v8f acc0 = {}, acc1 = {}, acc2 = {}, acc3 = {};
  #pragma unroll 4
  for (int k0 = 0; k0 < K; k0 += 4) {
    v2f a = *(const v2f*)(arow + k0);
    if (RELU_IN) { a.x = fmaxf(a.x, 0.0f); a.y = fmaxf(a.y, 0.0f); }
    const float* bk = bbase + (size_t)k0 * N;
    v2f b0, b1, b2, b3;
    b0.x = bk[0];  b0.y = bk[N];
    b1.x = bk[16]; b1.y = bk[N + 16];
    b2.x = bk[32]; b2.y = bk[N + 32];
    b3.x = bk[48]; b3.y = bk[N + 48];
    acc0 = __builtin_amdgcn_wmma_f32_16x16x4_f32(false, a, false, b0, (short)0, acc0, false, false);
    acc1 = __builtin_amdgcn_wmma_f32_16x16x4_f32(false, a, false, b1, (short)0, acc1, false, false);
    acc2 = __builtin_amdgcn_wmma_f32_16x16x4_f32(false, a, false, b2, (short)0, acc2, false, false);
    acc3 = __builtin_amdgcn_wmma_f32_16x16x4_f32(false, a, false, b3, (short)0, acc3, false, false);
  }

  float* cbase = C + (size_t)(mt * 16 + 8 * kh) * N + ns * 64 + mn;
  v8f accs[4] = {acc0, acc1, acc2, acc3};
  #pragma unroll
  for (int t = 0; t < 4; ++t) {
    float bv = BIAS ? bias[ns * 64 + 16 * t + mn] : 0.0f;
    #pragma unroll
    for (int v = 0; v < 8; ++v) {
      float x = accs[t][v] + bv;
      if (RELU_OUT) x = fmaxf(x, 0.0f);
      cbase[(size_t)v * N + 16 * t] = x;
    }
  }
}

// ---------------- host-side launch ----------------

static inline int cdiv(int a, int b) { return (a + b - 1) / b; }

extern "C" void kernel_launch(void* const* d_in, const int* in_sizes, int n_in,
                              void* d_out, int out_size, void* d_ws, size_t ws_size,
                              hipStream_t stream) {
  const float* x   = (const float*)d_in[0];
  const float* W1  = (const float*)d_in[1];
  const float* b1  = (const float*)d_in[2];
  const float* W2  = (const float*)d_in[3];
  const float* b2  = (const float*)d_in[4];
  const float* Wf1 = (const float*)d_in[5];
  const float* bf1 = (const float*)d_in[6];
  const float* Wf2 = (const float*)d_in[7];
  const float* bf2 = (const float*)d_in[8];
  const int* eidx  = (const int*)d_in[9];
  const int* batch = (const int*)d_in[10];

  const int hid     = in_sizes[2];             // 256
  const int fin     = in_sizes[1] / hid;       // 64
  const int nNodes  = in_sizes[0] / fin;       // 100000
  const int nE      = in_sizes[9] / 2;         // 1600000
  const int nGraphs = out_size / NCLS;         // 4096

  const int* src = eidx;
  const int* dst = eidx + nE;

  // workspace layout (floats)
  float* ws = (float*)d_ws;
  size_t o = 0;
  float* dinv = ws + o;  o += 102400;                       // >= nNodes
  float* bufA = ws + o;  o += (size_t)nNodes * hid;         // xw
  float* bufB = ws + o;  o += (size_t)nNodes * hid;         // agg / h
  float* pool = ws + o;  o += (size_t)nGraphs * hid;
  float* cnt  = ws + o;  o += nGraphs;
  float* fc1  = ws + o;  o += (size_t)nGraphs * hid;
  (void)ws_size; (void)n_in;

  const int nf = nNodes * hid;
  const int gf = nGraphs * hid;

  // 1) dinv = rsqrt(deg), deg includes self-loop
  fill_kernel<<<cdiv(nNodes, 256), 256, 0, stream>>>(dinv, 1.0f, nNodes);
  deg_count_kernel<<<cdiv(nE, 256), 256, 0, stream>>>(dst, dinv, nE);
  deg_rsqrt_kernel<<<cdiv(nNodes, 256), 256, 0, stream>>>(dinv, nNodes);

  // 2) layer 1: xw1 = x @ W1
  {
    int strips = (nNodes / 16) * (hid / 64);
    wmma_gemm_f32_strip_kernel<false, false, false>
        <<<cdiv(strips, 8), 256, 0, stream>>>(x, W1, nullptr, bufA, nNodes, hid, fin);
  }
  agg_init_kernel<<<cdiv(nf, 256), 256, 0, stream>>>(bufA, dinv, b1, bufB, nf);
  edge_scatter_kernel<<<cdiv(nE, 8), 256, 0, stream>>>(bufA, dinv, src, dst, bufB, nE);
  // (layer-1 ReLU fused into next GEMM's A-load)

  // 3) layer 2: xw2 = relu(h1) @ W2
  {
    int strips = (nNodes / 16) * (hid / 64);
    wmma_gemm_f32_strip_kernel<true, false, false>
        <<<cdiv(strips, 8), 256, 0, stream>>>(bufB, W2, nullptr, bufA, nNodes, hid, hid);
  }
  agg_init_kernel<<<cdiv(nf, 256), 256, 0, stream>>>(bufA, dinv, b2, bufB, nf);
  edge_scatter_kernel<<<cdiv(nE, 8), 256, 0, stream>>>(bufA, dinv, src, dst, bufB, nE);
  // (layer-2 ReLU fused into pool_acc)

  // 4) global mean pool of relu(h2)
  fill_kernel<<<cdiv(gf, 256), 256, 0, stream>>>(pool, 0.0f, gf);
  fill_kernel<<<cdiv(nGraphs, 256), 256, 0, stream>>>(cnt, 0.0f, nGraphs);
  pool_acc_kernel<<<cdiv(nNodes, 8), 256, 0, stream>>>(bufB, batch, pool, cnt, nNodes);
  pool_div_kernel<<<cdiv(gf, 256), 256, 0, stream>>>(pool, cnt, gf);

  // 5) head: relu(pooled @ Wf1 + bf1) @ Wf2 + bf2
  {
    int strips = (nGraphs / 16) * (hid / 64);
    wmma_gemm_f32_strip_kernel<false, true, true>
        <<<cdiv(strips, 8), 256, 0, stream>>>(pool, Wf1, bf1, fc1, nGraphs, hid, hid);
  }
  fc2_kernel<<<cdiv(nGraphs * NCLS, 256), 256, 0, stream>>>(fc1, Wf2, bf2, (float*)d_out, nGraphs);
}